// KGLinkPredictor_13907104104981
// MI455X (gfx1250) — compile-verified
//
#include <hip/hip_runtime.h>
#include <hip/hip_bf16.h>

// R-GCN (FastRGCNConv, mean aggregation) forward, 3 layers.
//   N = 50000 nodes, R = 16 relations, HID = 128, E = 1.6M edges.
// Per layer:
//   agg[n, r, :]  = mean over edges (src->n, rel r) of x[src]         (scatter)
//   out           = AGG[N, R*HID] @ Wflat[R*HID, HID]                 (WMMA f32 GEMM)
//                 + x @ root + bias,   ReLU between layers.

#define HID    128
#define RREL   16
#define KAGG   (RREL * HID)     // 2048
#define KTOT   (KAGG + HID)     // 2176  (agg part + root/self part)
#define KCHUNK 64

typedef float v2f __attribute__((ext_vector_type(2)));
typedef float v8f __attribute__((ext_vector_type(8)));

// ---------------------------------------------------------------- zero scratch
__global__ __launch_bounds__(256) void rgcn_zero(float* __restrict__ p, size_t count4)
{
    size_t i = ((size_t)blockIdx.x * 256 + threadIdx.x) * 4;
    if (i < count4) {
        float4 z = make_float4(0.f, 0.f, 0.f, 0.f);
        *(float4*)(p + i) = z;
    }
}

// ------------------------------------------------------- edge scatter (atomics)
// One wave32 per edge: lane loads float4 of x[src] (coalesced 512B row),
// atomically adds into agg[(tgt*R + rel)*HID + lane*4 ..].
__global__ __launch_bounds__(256) void rgcn_scatter(
    const int* __restrict__ src, const int* __restrict__ tgt,
    const int* __restrict__ etype, const float* __restrict__ x,
    float* __restrict__ agg, float* __restrict__ cnt, int nedges)
{
    int e = blockIdx.x * 8 + (threadIdx.x >> 5);
    if (e >= nedges) return;
    int lane = threadIdx.x & 31;

    int    s   = src[e];
    size_t seg = (size_t)tgt[e] * RREL + (size_t)etype[e];

    float4 v = *(const float4*)&x[(size_t)s * HID + lane * 4];
    float* dst = &agg[seg * HID + lane * 4];
    atomicAdd(dst + 0, v.x);
    atomicAdd(dst + 1, v.y);
    atomicAdd(dst + 2, v.z);
    atomicAdd(dst + 3, v.w);
    if (lane == 0) atomicAdd(&cnt[seg], 1.0f);
}

// ----------------------------------------------------------- WMMA GEMM + root
// Block: 16 rows x 128 cols of output. 8 waves, each owns one 16x16 tile.
// Unified K loop: k <  KAGG  -> A = agg[n][k] * 1/max(cnt,1),  B = Wflat[k][o]
//                 k >= KAGG  -> A = x[n][k-KAGG],               B = root[k-KAGG][o]
__global__ __launch_bounds__(256) void rgcn_gemm(
    const float* __restrict__ agg, const float* __restrict__ cnt,
    const float* __restrict__ x,   const float* __restrict__ Wl,
    const float* __restrict__ rootl, const float* __restrict__ biasl,
    float* __restrict__ out, int nrows, int relu)
{
    __shared__ float sA[16][KCHUNK + 4];        // +4 pad: avoids bank conflicts
    __shared__ float sB[KCHUNK][HID + 4];
    __shared__ float sInv[16][RREL];

    int m0 = blockIdx.x * 16;
    if (m0 >= nrows) return;                    // uniform per block
    int tid  = threadIdx.x;
    int lane = tid & 31;
    int wave = tid >> 5;                        // wave32: column tile n0 = wave*16
    int n0   = wave * 16;

    // per-(row, relation) reciprocal counts: 256 values, one per thread
    {
        int row = tid >> 4, r = tid & 15;
        float c = cnt[(size_t)(m0 + row) * RREL + r];
        sInv[row][r] = 1.0f / fmaxf(c, 1.0f);
    }
    __syncthreads();

    v8f acc = {};                               // 16x16 f32 accumulator (8 VGPRs)
    int colA    = lane & 15;                    // M (for A) / N (for B,D)
    int halfsel = (lane >> 4) << 1;             // lanes 16-31 hold K+2 half

    for (int k0 = 0; k0 < KTOT; k0 += KCHUNK) {
        // --- stage A chunk: 16 rows x 64 K (1024 elems, float4 per thread)
        {
            int idx  = tid * 4;
            int row  = idx >> 6;                // / KCHUNK
            int kk   = idx & (KCHUNK - 1);
            int gk   = k0 + kk;
            int grow = m0 + row;
            float4 v;
            if (gk < KAGG) {
                v = *(const float4*)&agg[(size_t)grow * KAGG + gk];
                float inv = sInv[row][gk >> 7];
                v.x *= inv; v.y *= inv; v.z *= inv; v.w *= inv;
            } else {
                v = *(const float4*)&x[(size_t)grow * HID + (gk - KAGG)];
            }
            *(float4*)&sA[row][kk] = v;
        }
        // --- stage B chunk: 64 K x 128 cols (8192 elems, 8 float4 per thread)
        for (int i = 0; i < 8; ++i) {
            int idx = (tid + i * 256) * 4;
            int kk  = idx >> 7;                 // / HID
            int cc  = idx & (HID - 1);
            int gk  = k0 + kk;
            const float* p = (gk < KAGG) ? &Wl[(size_t)gk * HID + cc]
                                         : &rootl[(size_t)(gk - KAGG) * HID + cc];
            *(float4*)&sB[kk][cc] = *(const float4*)p;
        }
        __syncthreads();

        // --- 16 WMMA steps of K=4 over this chunk
        for (int kk = 0; kk < KCHUNK; kk += 4) {
            int ak = kk + halfsel;
            v2f a, b;
            a.x = sA[colA][ak];
            a.y = sA[colA][ak + 1];
            b.x = sB[ak][n0 + colA];
            b.y = sB[ak + 1][n0 + colA];
            acc = __builtin_amdgcn_wmma_f32_16x16x4_f32(
                false, a, false, b, (short)0, acc, false, false);
        }
        __syncthreads();
    }

    // --- epilogue: + bias, optional ReLU, store D (VGPR v -> row v + 8*half)
    int   col   = n0 + colA;
    float bval  = biasl[col];
    int   rbase = m0 + ((lane >> 4) << 3);
    #pragma unroll
    for (int v = 0; v < 8; ++v) {
        float o = acc[v] + bval;
        if (relu) o = fmaxf(o, 0.0f);
        out[(size_t)(rbase + v) * HID + col] = o;
    }
}

// --------------------------------------------------------------------- driver
extern "C" void kernel_launch(void* const* d_in, const int* in_sizes, int n_in,
                              void* d_out, int out_size, void* d_ws, size_t ws_size,
                              hipStream_t stream)
{
    const int*   edge_index = (const int*)d_in[0];   // [2, E]
    const int*   etype      = (const int*)d_in[1];   // [E]
    const float* node_init  = (const float*)d_in[2]; // [N, 128]
    const float* W          = (const float*)d_in[3]; // [3, 16, 128, 128]
    const float* root       = (const float*)d_in[4]; // [3, 128, 128]
    const float* bias       = (const float*)d_in[5]; // [3, 128]

    int E = in_sizes[1];
    int N = in_sizes[2] / HID;
    const int* src = edge_index;
    const int* tgt = edge_index + E;

    float*  ws   = (float*)d_ws;
    size_t  aggN = (size_t)N * RREL * HID;           // 102.4M floats
    size_t  cntN = (size_t)N * RREL;                 // 800K floats
    float*  agg  = ws;
    float*  cnt  = ws + aggN;                        // contiguous with agg (zeroed together)
    float*  xb0  = cnt + cntN;
    float*  xb1  = xb0 + (size_t)N * HID;
    float*  outF = (float*)d_out;

    size_t zcount  = aggN + cntN;                    // multiple of 4
    int    zblocks = (int)((zcount / 4 + 255) / 256);
    int    eblocks = (E + 7) / 8;
    int    gblocks = (N + 15) / 16;                  // N = 50000 -> exact tiles

    const float* xin = node_init;
    for (int layer = 0; layer < 3; ++layer) {
        float* xout = (layer == 0) ? xb0 : (layer == 1) ? xb1 : outF;

        rgcn_zero<<<zblocks, 256, 0, stream>>>(agg, zcount);
        rgcn_scatter<<<eblocks, 256, 0, stream>>>(src, tgt, etype, xin, agg, cnt, E);
        rgcn_gemm<<<gblocks, 256, 0, stream>>>(
            agg, cnt, xin,
            W    + (size_t)layer * RREL * HID * HID,
            root + (size_t)layer * HID * HID,
            bias + (size_t)layer * HID,
            xout, N, layer < 2 ? 1 : 0);

        xin = xout;
    }
}